// le_mse_72181220376973
// MI455X (gfx1250) — compile-verified
//
#include <hip/hip_runtime.h>
#include <math.h>

typedef __attribute__((ext_vector_type(2))) float v2f;
typedef __attribute__((ext_vector_type(8))) float v8f;

// ---------------------------------------------------------------------------
// Closed-form matrix log of a symmetric 3x3 SPD matrix.
// Eigenvalues via the trigonometric method; logm via Cayley-Hamilton:
//   logm(A) = c0*I + c1*A + c2*A^2,  c = Lagrange interp of log at eigenvalues.
// Output L = {l11, l22, l33, l12, l13, l23}.
// ---------------------------------------------------------------------------
__device__ __forceinline__ void spd_logm(float a11, float a22, float a33,
                                         float a12, float a13, float a23,
                                         float L[6]) {
    const float third = 0.3333333333333333f;
    float q  = (a11 + a22 + a33) * third;
    float b11 = a11 - q, b22 = a22 - q, b33 = a33 - q;
    float p1 = a12 * a12 + a13 * a13 + a23 * a23;
    float p2 = b11 * b11 + b22 * b22 + b33 * b33 + 2.0f * p1;
    float p  = sqrtf(p2 * (1.0f / 6.0f)) + 1e-30f;
    float invp = 1.0f / p;
    // det(A - qI)
    float det = b11 * (b22 * b33 - a23 * a23)
              - a12 * (a12 * b33 - a23 * a13)
              + a13 * (a12 * a23 - b22 * a13);
    float r = 0.5f * det * invp * invp * invp;
    r = fminf(1.0f, fmaxf(-1.0f, r));
    float phi  = acosf(r) * third;
    float twop = 2.0f * p;
    float e1 = q + twop * __cosf(phi);
    float e3 = q + twop * __cosf(phi + 2.0943951023931953f); // +2*pi/3
    float e2 = 3.0f * q - e1 - e3;

    // Guarded pairwise differences (eigenvalues generically distinct)
    float d12 = e1 - e2, d13 = e1 - e3, d23 = e2 - e3;
    d12 = (fabsf(d12) < 1e-12f) ? 1e-12f : d12;
    d13 = (fabsf(d13) < 1e-12f) ? 1e-12f : d13;
    d23 = (fabsf(d23) < 1e-12f) ? 1e-12f : d23;

    float l1 = __logf(fmaxf(e1, 1e-30f));
    float l2 = __logf(fmaxf(e2, 1e-30f));
    float l3 = __logf(fmaxf(e3, 1e-30f));

    // Lagrange weights: (e1-e2)(e1-e3)=d12*d13, (e2-e1)(e2-e3)=-d12*d23,
    //                   (e3-e1)(e3-e2)=d13*d23
    float w1 =  l1 / (d12 * d13);
    float w2 = -l2 / (d12 * d23);
    float w3 =  l3 / (d13 * d23);

    float c2 = w1 + w2 + w3;
    float c1 = -(w1 * (e2 + e3) + w2 * (e1 + e3) + w3 * (e1 + e2));
    float c0 = w1 * e2 * e3 + w2 * e1 * e3 + w3 * e1 * e2;

    // A^2 (symmetric)
    float s11 = a11 * a11 + a12 * a12 + a13 * a13;
    float s22 = a12 * a12 + a22 * a22 + a23 * a23;
    float s33 = a13 * a13 + a23 * a23 + a33 * a33;
    float s12 = a11 * a12 + a12 * a22 + a13 * a23;
    float s13 = a11 * a13 + a12 * a23 + a13 * a33;
    float s23 = a12 * a13 + a22 * a23 + a23 * a33;

    L[0] = c0 + c1 * a11 + c2 * s11;
    L[1] = c0 + c1 * a22 + c2 * s22;
    L[2] = c0 + c1 * a33 + c2 * s33;
    L[3] = c1 * a12 + c2 * s12;
    L[4] = c1 * a13 + c2 * s13;
    L[5] = c1 * a23 + c2 * s23;
}

// ---------------------------------------------------------------------------
// Block-wide (256-thread) sum using V_WMMA_F32_16X16X4_F32.
// Wave 0 loads all 256 partials as four 16x4 f32 A-matrices and multiplies by
// a ones B-matrix, accumulating in C: each WMMA reduces 64 values along K.
// D[m][n] is constant over n; summing the 8 C VGPRs + one shfl_xor(16)
// yields the full 256-value sum in every lane of wave 0.
// Caller must pass a 256-entry LDS buffer; requires blockDim.x == 256.
// ---------------------------------------------------------------------------
__device__ __forceinline__ float block_reduce_wmma(float s, float* sdata) {
    const int tid = threadIdx.x;
    sdata[tid] = s;
    __syncthreads();
    float r = 0.0f;
    if (tid < 32) {  // wave-uniform branch: wave 0 fully active (EXEC all 1s)
        v2f ones = {1.0f, 1.0f};
        v8f c = {};
#pragma unroll
        for (int g = 0; g < 4; ++g) {
            v2f a = {sdata[g * 64 + tid], sdata[g * 64 + tid + 32]};
            c = __builtin_amdgcn_wmma_f32_16x16x4_f32(
                /*neg_a=*/false, a, /*neg_b=*/false, ones,
                /*c_mod=*/(short)0, c, /*reuse_a=*/false, /*reuse_b=*/false);
        }
        r = c[0] + c[1] + c[2] + c[3] + c[4] + c[5] + c[6] + c[7];
        r += __shfl_xor(r, 16, 32);
    }
    return r;  // valid in all lanes of wave 0
}

__global__ void __launch_bounds__(256)
le_partial_kernel(const float* __restrict__ D1, const float* __restrict__ D2,
                  float* __restrict__ partial, int n) {
    const int tid = threadIdx.x;
    float s = 0.0f;
    long long stride = (long long)gridDim.x * 256;
    for (long long i = (long long)blockIdx.x * 256 + tid; i < n; i += stride) {
        const float* g1 = D1 + 9 * i;
        const float* g2 = D2 + 9 * i;
        // Symmetric input: read 6 of 9 elements
        float L1[6], L2[6];
        spd_logm(g1[0], g1[4], g1[8], g1[1], g1[2], g1[5], L1);
        spd_logm(g2[0], g2[4], g2[8], g2[1], g2[2], g2[5], L2);
        float d0 = L1[0] - L2[0], d1 = L1[1] - L2[1], d2 = L1[2] - L2[2];
        float d3 = L1[3] - L2[3], d4 = L1[4] - L2[4], d5 = L1[5] - L2[5];
        s += d0 * d0 + d1 * d1 + d2 * d2
           + 2.0f * (d3 * d3 + d4 * d4 + d5 * d5);
    }
    __shared__ float sdata[256];
    float r = block_reduce_wmma(s, sdata);
    if (tid == 0) partial[blockIdx.x] = r;
}

__global__ void __launch_bounds__(256)
le_final_kernel(const float* __restrict__ partial, int nb,
                float* __restrict__ out, float invN) {
    const int tid = threadIdx.x;
    float s = 0.0f;
    for (int i = tid; i < nb; i += 256) s += partial[i];
    __shared__ float sdata[256];
    float r = block_reduce_wmma(s, sdata);
    if (tid == 0) out[0] = r * invN;
}

extern "C" void kernel_launch(void* const* d_in, const int* in_sizes, int n_in,
                              void* d_out, int out_size, void* d_ws, size_t ws_size,
                              hipStream_t stream) {
    const float* D1 = (const float*)d_in[0];
    const float* D2 = (const float*)d_in[1];
    float* partial  = (float*)d_ws;
    float* out      = (float*)d_out;

    const int n = in_sizes[0] / 9;  // number of 3x3 matrices
    int nb = (n + 255) / 256;
    if (nb > 1024) nb = 1024;       // 1024 blocks * 8 waves = 8192 wave32s
    if (nb < 1) nb = 1;

    le_partial_kernel<<<nb, 256, 0, stream>>>(D1, D2, partial, n);
    le_final_kernel<<<1, 256, 0, stream>>>(partial, nb, out, 1.0f / (float)n);
}